// DistributedEmbedding_13726715478733
// MI455X (gfx1250) — compile-verified
//
#include <hip/hip_runtime.h>

typedef float v2f __attribute__((ext_vector_type(2)));
typedef float v8f __attribute__((ext_vector_type(8)));

#define BATCH      4096
#define SLOT_NUM   26
#define EMB_DIM    64
#define N_ROWS     (BATCH * SLOT_NUM)     // 106496
#define NNZ        532480
#define TILE_M     16
#define N_TILES    (N_ROWS / TILE_M)      // 6656 (exact)
#define WAVES_PB   8                      // 256 threads = 8 wave32

// One 4-nnz chunk: Out_tile += A[16x4] x B[4x64] via 4 WMMA f32 16x16x4 (one per 16-col N-tile).
// r0..r3: row offsets within tile (-1 => weight 0). v0..v3: embedding keys (clamped-safe).
__device__ __forceinline__ void chunk_mma(int m, int hiHalf, const float* __restrict__ embm,
                                          int r0, int r1, int r2, int r3,
                                          long long v0, long long v1, long long v2, long long v3,
                                          v8f& acc0, v8f& acc1, v8f& acc2, v8f& acc3)
{
    // A (16x4 f32): lanes0-15: {K0,K1}, lanes16-31: {K2,K3}; A[m][k] = (nnz k lands on row m)
    int ra0 = hiHalf ? r2 : r0;
    int ra1 = hiHalf ? r3 : r1;
    v2f a;
    a.x = (ra0 == m) ? 1.0f : 0.0f;
    a.y = (ra1 == m) ? 1.0f : 0.0f;

    // B (4x16 f32) per N-tile: VGPR0 = {K0 | K2}, VGPR1 = {K1 | K3}, N = lane&15
    long long iLo = hiHalf ? v2 : v0;
    long long iHi = hiHalf ? v3 : v1;
    const float* pLo = embm + iLo * EMB_DIM;   // coalesced 64B row-quarters
    const float* pHi = embm + iHi * EMB_DIM;

    v2f b0, b1, b2, b3;
    b0.x = pLo[0];   b0.y = pHi[0];
    b1.x = pLo[16];  b1.y = pHi[16];
    b2.x = pLo[32];  b2.y = pHi[32];
    b3.x = pLo[48];  b3.y = pHi[48];

    acc0 = __builtin_amdgcn_wmma_f32_16x16x4_f32(false, a, false, b0, (short)0, acc0, false, false);
    acc1 = __builtin_amdgcn_wmma_f32_16x16x4_f32(false, a, false, b1, (short)0, acc1, false, false);
    acc2 = __builtin_amdgcn_wmma_f32_16x16x4_f32(false, a, false, b2, (short)0, acc2, false, false);
    acc3 = __builtin_amdgcn_wmma_f32_16x16x4_f32(false, a, false, b3, (short)0, acc3, false, false);
}

__global__ __launch_bounds__(256)
void DistributedEmbedding_wmma_kernel(const int*   __restrict__ values,
                                      const int*   __restrict__ rows,
                                      const float* __restrict__ emb,
                                      float*       __restrict__ out)
{
    const int lane   = threadIdx.x & 31;
    const int wave   = threadIdx.x >> 5;
    const int tile   = blockIdx.x * WAVES_PB + wave;
    const int base   = tile * TILE_M;
    const int m      = lane & 15;          // column within N-tile / row within half
    const int hiHalf = (lane >> 4) & 1;    // 0: lanes 0-15, 1: lanes 16-31

    // ---- 17 lower_bound searches in parallel: lane l -> lb(base + min(l,16)) ----
    int target = base + (lane < 16 ? lane : 16);
    int lo = 0, hi = NNZ;
    while (lo < hi) {
        int mid = (lo + hi) >> 1;
        if (rows[mid] < target) lo = mid + 1; else hi = mid;
    }
    // segment bounds (scalarized: chunk loop is a uniform s_cbranch, EXEC stays all-1s for WMMA)
    int seg_start = __builtin_amdgcn_readfirstlane(lo);
    int se_v      = __shfl(lo, 16, 32);
    int seg_end   = __builtin_amdgcn_readfirstlane(se_v);

    // per-row count -> 1/count (valid in lanes 0-15, duplicated in 16-31)
    int cnt   = __shfl(lo, m + 1, 32) - __shfl(lo, m, 32);
    float inv = 1.0f / fmaxf((float)cnt, 1.0f);

    v8f acc0 = {}, acc1 = {}, acc2 = {}, acc3 = {};

    const float* embm = emb + m;

    // ---- main loop: full 4-nnz chunks, branch-free body ----
    const int kfull_end = seg_start + ((seg_end - seg_start) & ~3);
    for (int k = seg_start; k < kfull_end; k += 4) {
        int r0 = rows[k]     - base;
        int r1 = rows[k + 1] - base;
        int r2 = rows[k + 2] - base;
        int r3 = rows[k + 3] - base;
        long long v0 = values[k];
        long long v1 = values[k + 1];
        long long v2 = values[k + 2];
        long long v3 = values[k + 3];
        chunk_mma(m, hiHalf, embm, r0, r1, r2, r3, v0, v1, v2, v3, acc0, acc1, acc2, acc3);
    }

    // ---- tail chunk: 1-3 entries, clamped (invalid weight 0, key clamped in-range) ----
    if (kfull_end < seg_end) {
        const int k = kfull_end, last = seg_end - 1;
        int r0 = rows[k] - base;
        int r1 = (k + 1 <= last) ? rows[k + 1] - base : -1;
        int r2 = (k + 2 <= last) ? rows[k + 2] - base : -1;
        int r3 = -1;                                   // tail has at most 3 entries
        long long v0 = values[k];
        long long v1 = values[(k + 1 <= last) ? k + 1 : last];
        long long v2 = values[(k + 2 <= last) ? k + 2 : last];
        long long v3 = v0;
        chunk_mma(m, hiHalf, embm, r0, r1, r2, r3, v0, v1, v2, v3, acc0, acc1, acc2, acc3);
    }

    // D layout: VGPR i -> M=i (lanes0-15) / M=8+i (lanes16-31), N=lane&15.
    // Apply mean scale (1/count of row M) and store; each output element written exactly once.
    #pragma unroll
    for (int i = 0; i < 8; ++i) {
        int mrow  = i + (hiHalf << 3);
        float s   = __shfl(inv, mrow, 32);          // inv lives in lanes 0-15
        long long o = (long long)(base + mrow) * EMB_DIM + m;
        out[o +  0] = acc0[i] * s;
        out[o + 16] = acc1[i] * s;
        out[o + 32] = acc2[i] * s;
        out[o + 48] = acc3[i] * s;
    }
}

extern "C" void kernel_launch(void* const* d_in, const int* in_sizes, int n_in,
                              void* d_out, int out_size, void* d_ws, size_t ws_size,
                              hipStream_t stream) {
    const int*   values = (const int*)d_in[0];
    const int*   rows   = (const int*)d_in[1];
    const float* emb    = (const float*)d_in[2];
    float*       out    = (float*)d_out;

    dim3 grid(N_TILES / WAVES_PB);   // 832 blocks x 8 waves = 6656 tiles (exact cover)
    DistributedEmbedding_wmma_kernel<<<grid, 256, 0, stream>>>(values, rows, emb, out);
}